// GELU159_39857296507312
// MI455X (gfx1250) — compile-verified
//
#include <hip/hip_runtime.h>
#include <math.h>

typedef float v2f __attribute__((ext_vector_type(2)));
typedef float v4f __attribute__((ext_vector_type(4)));
typedef float v8f __attribute__((ext_vector_type(8)));

#define HG   16
#define TT   4096
#define BB   4
#define DD   2048
#define ROWS (BB * TT)           // 16384

// Fast reciprocal: v_rcp_f32 (~1 ulp), avoids IEEE div expansion (div_scale/div_fmas)
__device__ __forceinline__ float frcp(float v) {
    return __builtin_amdgcn_rcpf(v);
}
__device__ __forceinline__ float sigm(float v) {
    return frcp(1.0f + __expf(-v));
}
__device__ __forceinline__ float tanh_safe(float v) {
    float ax = fabsf(v);
    float e  = __expf(-2.0f * ax);            // in (0,1], no overflow
    float t  = (1.0f - e) * frcp(1.0f + e);
    return v < 0.0f ? -t : t;
}
__device__ __forceinline__ float gelu_tanh(float x) {
    const float k0 = 0.7978845608028654f;     // sqrt(2/pi)
    const float c  = 0.044715f;
    float u = k0 * (x + c * x * x * x);
    return 0.5f * x * (1.0f + tanh_safe(u));
}

// ---------------- kernel 1: inv_std ----------------
__global__ void k_prep(const float* __restrict__ mu, const float* __restrict__ sq,
                       float* __restrict__ inv) {
    int d = blockIdx.x * blockDim.x + threadIdx.x;
    if (d < DD) {
        float m = mu[d];
        float v = sq[d] - m * m;
        v = fmaxf(v, 1e-4f);
        inv[d] = frcp(sqrtf(v) + 1e-5f);
    }
}

// ---------------- kernel 2: surprise = mean |z| over D ----------------
__global__ void k_surp(const float* __restrict__ x, const float* __restrict__ mu,
                       const float* __restrict__ inv, float* __restrict__ surp) {
    int row = blockIdx.x;                 // (b,t) row
    int tid = threadIdx.x;                // 256 threads
    const v4f* x4 = (const v4f*)(x + (size_t)row * DD);
    const v4f* m4 = (const v4f*)mu;
    const v4f* i4 = (const v4f*)inv;
    float acc = 0.0f;
#pragma unroll
    for (int i = 0; i < 2; ++i) {
        int idx = tid + i * 256;
        v4f xv = x4[idx];
        v4f mv = m4[idx];
        v4f iv = i4[idx];
        acc += fabsf((xv.x - mv.x) * iv.x);
        acc += fabsf((xv.y - mv.y) * iv.y);
        acc += fabsf((xv.z - mv.z) * iv.z);
        acc += fabsf((xv.w - mv.w) * iv.w);
    }
#pragma unroll
    for (int o = 16; o > 0; o >>= 1) acc += __shfl_down(acc, o, 32);
    __shared__ float red[8];
    if ((tid & 31) == 0) red[tid >> 5] = acc;
    __syncthreads();
    if (tid == 0) {
        float s = 0.0f;
#pragma unroll
        for (int w = 0; w < 8; ++w) s += red[w];
        surp[row] = s * (1.0f / (float)DD);
    }
}

// ---------------- kernel 3: serial GRU scan, one wave per batch ----------------
// Lanes 0..15 compute r-rows, lanes 16..31 z-rows, all lanes redundantly
// compute the n-row for j = lane & 15. h replicated across both halves.
__global__ void k_gru(const float* __restrict__ surp,
                      const float* __restrict__ w_ih, const float* __restrict__ w_hh,
                      const float* __restrict__ b_ih, const float* __restrict__ b_hh,
                      float* __restrict__ gru_out) {
    int batch = blockIdx.x;               // 4 blocks of 32
    int lane  = threadIdx.x;              // 0..31
    int j     = lane & 15;

    // rows 0..15 = whr, 16..31 = whz, 32..47 = whn
    float wrz[16], wn[16];
#pragma unroll
    for (int k = 0; k < 16; ++k) wrz[k] = w_hh[lane * 16 + k];
#pragma unroll
    for (int k = 0; k < 16; ++k) wn[k] = w_hh[(32 + j) * 16 + k];
    float brz   = b_hh[lane];
    float bn    = b_hh[32 + j];
    float wi_rz = w_ih[lane];
    float bi_rz = b_ih[lane];
    float wi_n  = w_ih[32 + j];
    float bi_n  = b_ih[32 + j];

    const float* sp = surp + batch * TT;
    float h = 0.0f;
    float s_cur = sp[0];

    for (int t = 0; t < TT; ++t) {
        int tn = (t + 1 < TT) ? (t + 1) : (TT - 1);
        float s_nxt = sp[tn];             // double-buffered, off critical path

        // h @ W.T, two independent chains of 8 per gate pair
        float a0 = brz, a1 = 0.0f;
        float n0 = bn,  n1 = 0.0f;
#pragma unroll
        for (int k = 0; k < 8; ++k) {
            float hk = __shfl(h, k, 32);
            a0 = fmaf(wrz[k], hk, a0);
            n0 = fmaf(wn[k],  hk, n0);
        }
#pragma unroll
        for (int k = 8; k < 16; ++k) {
            float hk = __shfl(h, k, 32);
            a1 = fmaf(wrz[k], hk, a1);
            n1 = fmaf(wn[k],  hk, n1);
        }
        float acc_rz = a0 + a1;
        float acc_n  = n0 + n1;

        float i_rz = fmaf(wi_rz, s_cur, bi_rz);
        float rz   = sigm(i_rz + acc_rz);       // r on lanes<16, z on lanes>=16
        float r    = __shfl(rz, j, 32);
        float z    = __shfl(rz, 16 + j, 32);
        float i_n  = fmaf(wi_n, s_cur, bi_n);
        float n    = tanh_safe(fmaf(r, acc_n, i_n));
        h = fmaf(z, h - n, n);                  // (1-z)*n + z*h

        if (lane < 16)
            gru_out[((size_t)(batch * TT + t)) * HG + lane] = h;
        s_cur = s_nxt;
    }
}

// ---------------- kernel 4: gate via WMMA f32 16x16x4 ----------------
// Each wave: 16 rows of gru_out times w_lin (broadcast across N), K=16 as 4 chunks.
__global__ void k_gate(const float* __restrict__ gru_out,
                       const float* __restrict__ w_lin, const float* __restrict__ b_lin,
                       const float* __restrict__ log_alpha, float* __restrict__ gate) {
    int gtid   = blockIdx.x * blockDim.x + threadIdx.x;
    int waveId = gtid >> 5;               // 0..1023, 16 rows each
    int lane   = threadIdx.x & 31;
    int half   = lane >> 4;
    int m      = lane & 15;
    int row    = waveId * 16 + m;

    float alpha = __expf(log_alpha[0]);
    float bl    = b_lin[0];

    v8f acc = {};
#pragma unroll
    for (int c = 0; c < 4; ++c) {
        int col = 4 * c + 2 * half;       // A layout: K = vgpr + 2*half
        v2f a, b;
        a.x = gru_out[(size_t)row * HG + col];
        a.y = gru_out[(size_t)row * HG + col + 1];
        b.x = w_lin[col];                 // broadcast across all N columns
        b.y = w_lin[col + 1];
        acc = __builtin_amdgcn_wmma_f32_16x16x4_f32(
            false, a, false, b, (short)0, acc, false, false);
    }

    // D layout: VGPR v, lane l -> M = v + 8*(l>=16), N = l&15. Take column N = 0.
    if (m == 0) {
#pragma unroll
        for (int v = 0; v < 8; ++v) {
            float gr = acc[v] + bl;
            float g  = 1.0f + alpha * sigm(gr);
            gate[waveId * 16 + v + 8 * half] = g;
        }
    }
}

// ---------------- kernel 5: out = gelu(x) * gate, NT stores ----------------
__global__ void k_out(const float* __restrict__ x, const float* __restrict__ gate,
                      float* __restrict__ out) {
    int i = blockIdx.x * blockDim.x + threadIdx.x;   // float4 index, 512 per row
    const v4f* x4 = (const v4f*)x;
    v4f v = x4[i];
    float g = gate[i >> 9];
    v4f r;
    r.x = gelu_tanh(v.x) * g;
    r.y = gelu_tanh(v.y) * g;
    r.z = gelu_tanh(v.z) * g;
    r.w = gelu_tanh(v.w) * g;
    __builtin_nontemporal_store(r, ((v4f*)out) + i);  // keep x resident in 192MB L2
}

extern "C" void kernel_launch(void* const* d_in, const int* in_sizes, int n_in,
                              void* d_out, int out_size, void* d_ws, size_t ws_size,
                              hipStream_t stream) {
    const float* x         = (const float*)d_in[0];
    const float* w_ih      = (const float*)d_in[1];
    const float* w_hh      = (const float*)d_in[2];
    const float* b_ih      = (const float*)d_in[3];
    const float* b_hh      = (const float*)d_in[4];
    const float* w_lin     = (const float*)d_in[5];
    const float* b_lin     = (const float*)d_in[6];
    const float* log_alpha = (const float*)d_in[7];
    const float* ema_mean  = (const float*)d_in[8];
    const float* ema_sq    = (const float*)d_in[9];
    float* out = (float*)d_out;

    char* ws = (char*)d_ws;
    float* inv   = (float*)ws;                                  // 2048 f
    float* surp  = (float*)(ws + 8192);                         // 16384 f
    float* gruo  = (float*)(ws + 8192 + 65536);                 // 16384*16 f
    float* gate  = (float*)(ws + 8192 + 65536 + (size_t)ROWS * HG * 4); // 16384 f

    k_prep<<<8, 256, 0, stream>>>(ema_mean, ema_sq, inv);
    k_surp<<<ROWS, 256, 0, stream>>>(x, ema_mean, inv, surp);
    k_gru<<<BB, 32, 0, stream>>>(surp, w_ih, w_hh, b_ih, b_hh, gruo);
    k_gate<<<(ROWS / 16) * 32 / 256, 256, 0, stream>>>(gruo, w_lin, b_lin, log_alpha, gate);
    k_out<<<(ROWS * DD / 4) / 256, 256, 0, stream>>>(x, gate, out);
}